// SeModule1d_60679297958461
// MI455X (gfx1250) — compile-verified
//
#include <hip/hip_runtime.h>
#include <hip/hip_bf16.h>
#include <math.h>

// ---------------------------------------------------------------------------
// SE module for x:[B=16, C=256, L=16384] f32.
//   pool:   s[b,c] = mean_L x          (256 MB read  -> BW bound)
//   se_mlp: gate = sigmoid(BN(GELU(BN(s@w1^T))@w2^T))   (tiny; 1 wave, WMMA f32)
//   scale:  out = x * gate             (256 MB read + 256 MB write -> BW bound)
// Roofline: ~768 MB @ 23.3 TB/s ~= 33 us; matmuls are ~2 MFLOP (free).
// ---------------------------------------------------------------------------

typedef __attribute__((ext_vector_type(2))) float v2f;
typedef __attribute__((ext_vector_type(8))) float v8f;

#define B_DIM 16
#define C_DIM 256
#define CR_DIM 128
#define L_DIM 16384

// ---------------------------------------------------------------------------
// Kernel 1: global average pool over L. One workgroup per (b,c) row.
// ---------------------------------------------------------------------------
__global__ __launch_bounds__(256)
void se_pool_kernel(const float* __restrict__ x, float* __restrict__ s) {
    const int row = blockIdx.x;                       // 0 .. B*C-1
    const float4* xr = (const float4*)(x + (size_t)row * L_DIM);
    float sum = 0.f;
    for (int i = threadIdx.x; i < L_DIM / 4; i += 256) {
        float4 v = xr[i];                              // global_load_b128
        sum += (v.x + v.y) + (v.z + v.w);
    }
    // wave32 reduction
    #pragma unroll
    for (int off = 16; off > 0; off >>= 1)
        sum += __shfl_down(sum, off, 32);
    __shared__ float red[8];
    if ((threadIdx.x & 31) == 0) red[threadIdx.x >> 5] = sum;
    __syncthreads();
    if (threadIdx.x == 0) {
        float t = 0.f;
        #pragma unroll
        for (int w = 0; w < 8; ++w) t += red[w];
        s[row] = t * (1.0f / (float)L_DIM);
    }
}

// ---------------------------------------------------------------------------
// Kernel 2: the whole SE MLP in one wave using V_WMMA_F32_16X16X4_F32.
//   A (16x4 f32):  lane l holds row M=l%16, K = 2*(l/16)+v      (v = vgpr 0/1)
//   B (4x16 f32):  lane l holds col N=l%16, K = 2*(l/16)+v
//   C/D (16x16):   lane l holds col N=l%16, vgpr i -> row M = i + 8*(l/16)
// ---------------------------------------------------------------------------
__global__ __launch_bounds__(32)
void se_mlp_kernel(const float* __restrict__ s,     // [16,256]
                   const float* __restrict__ w1,    // [128,256]
                   const float* __restrict__ w2,    // [256,128]
                   const float* __restrict__ g1, const float* __restrict__ b1,
                   const float* __restrict__ g2, const float* __restrict__ b2,
                   float* __restrict__ gate) {      // [16,256]
    __shared__ float h1[B_DIM][CR_DIM];   // squeeze activations
    __shared__ float h2[B_DIM][C_DIM];    // excite pre-activations

    const int lane = threadIdx.x;         // 0..31
    const int col  = lane & 15;           // N (and M for A-frag)
    const int hi   = lane >> 4;           // K half selector

    // ---- h1 = s @ w1^T : 8 tiles of 16x16, K=256 in steps of 4 ----
    for (int n = 0; n < CR_DIM / 16; ++n) {
        v8f acc = {};
        const int nrow = n * 16 + col;    // w1 row index (output channel)
        for (int k = 0; k < C_DIM; k += 4) {
            const int k0 = k + 2 * hi;
            v2f a, bb;
            a.x  = s[col * C_DIM + k0 + 0];
            a.y  = s[col * C_DIM + k0 + 1];
            bb.x = w1[nrow * C_DIM + k0 + 0];   // B[k][n] = w1[n][k]
            bb.y = w1[nrow * C_DIM + k0 + 1];
            acc = __builtin_amdgcn_wmma_f32_16x16x4_f32(
                false, a, false, bb, (short)0, acc, false, false);
        }
        #pragma unroll
        for (int i = 0; i < 8; ++i)
            h1[i + 8 * hi][n * 16 + col] = acc[i];
    }
    __syncthreads();

    // ---- BN over batch (biased var) + exact-erf GELU; 4 cols per lane ----
    for (int c = lane * 4; c < lane * 4 + 4; ++c) {
        float mu = 0.f;
        #pragma unroll
        for (int m = 0; m < B_DIM; ++m) mu += h1[m][c];
        mu *= (1.f / B_DIM);
        float var = 0.f;
        #pragma unroll
        for (int m = 0; m < B_DIM; ++m) { float d = h1[m][c] - mu; var += d * d; }
        var *= (1.f / B_DIM);
        const float inv = rsqrtf(var + 1e-5f);
        const float ga = g1[c], be = b1[c];
        #pragma unroll
        for (int m = 0; m < B_DIM; ++m) {
            float v = (h1[m][c] - mu) * inv * ga + be;
            h1[m][c] = 0.5f * v * (1.f + erff(v * 0.70710678118654752f));
        }
    }
    __syncthreads();

    // ---- h2 = gelu(h1) @ w2^T : 16 tiles, K=128 ----
    for (int n = 0; n < C_DIM / 16; ++n) {
        v8f acc = {};
        const int nrow = n * 16 + col;    // w2 row index
        for (int k = 0; k < CR_DIM; k += 4) {
            const int k0 = k + 2 * hi;
            v2f a, bb;
            a.x  = h1[col][k0 + 0];
            a.y  = h1[col][k0 + 1];
            bb.x = w2[nrow * CR_DIM + k0 + 0];  // B[k][n] = w2[n][k]
            bb.y = w2[nrow * CR_DIM + k0 + 1];
            acc = __builtin_amdgcn_wmma_f32_16x16x4_f32(
                false, a, false, bb, (short)0, acc, false, false);
        }
        #pragma unroll
        for (int i = 0; i < 8; ++i)
            h2[i + 8 * hi][n * 16 + col] = acc[i];
    }
    __syncthreads();

    // ---- BN over batch + sigmoid -> gate; 8 cols per lane ----
    for (int c = lane * 8; c < lane * 8 + 8; ++c) {
        float mu = 0.f;
        #pragma unroll
        for (int m = 0; m < B_DIM; ++m) mu += h2[m][c];
        mu *= (1.f / B_DIM);
        float var = 0.f;
        #pragma unroll
        for (int m = 0; m < B_DIM; ++m) { float d = h2[m][c] - mu; var += d * d; }
        var *= (1.f / B_DIM);
        const float inv = rsqrtf(var + 1e-5f);
        const float ga = g2[c], be = b2[c];
        #pragma unroll
        for (int m = 0; m < B_DIM; ++m) {
            float v = (h2[m][c] - mu) * inv * ga + be;
            gate[m * C_DIM + c] = 1.f / (1.f + __expf(-v));
        }
    }
}

// ---------------------------------------------------------------------------
// Kernel 3: out = x * gate[b,c], float4-vectorized grid-stride stream.
// ---------------------------------------------------------------------------
__global__ __launch_bounds__(256)
void se_scale_kernel(const float* __restrict__ x, const float* __restrict__ gate,
                     float* __restrict__ out) {
    const long long total = (long long)B_DIM * C_DIM * L_DIM / 4;   // float4 count
    const long long stride = (long long)gridDim.x * 256;
    for (long long i = (long long)blockIdx.x * 256 + threadIdx.x; i < total; i += stride) {
        const int row = (int)(i >> 12);          // (i*4)/L_DIM : (b*C + c)
        const float g = gate[row];
        float4 v = ((const float4*)x)[i];        // global_load_b128
        v.x *= g; v.y *= g; v.z *= g; v.w *= g;
        ((float4*)out)[i] = v;                   // global_store_b128
    }
}

// ---------------------------------------------------------------------------
extern "C" void kernel_launch(void* const* d_in, const int* in_sizes, int n_in,
                              void* d_out, int out_size, void* d_ws, size_t ws_size,
                              hipStream_t stream) {
    (void)in_sizes; (void)n_in; (void)out_size; (void)ws_size;
    const float* x  = (const float*)d_in[0];   // [16,256,16384]
    const float* w1 = (const float*)d_in[1];   // [128,256]
    const float* g1 = (const float*)d_in[2];
    const float* b1 = (const float*)d_in[3];
    const float* w2 = (const float*)d_in[4];   // [256,128]
    const float* g2 = (const float*)d_in[5];
    const float* b2 = (const float*)d_in[6];
    float* out  = (float*)d_out;

    float* s    = (float*)d_ws;                // [16*256]
    float* gate = s + B_DIM * C_DIM;           // [16*256]

    se_pool_kernel<<<B_DIM * C_DIM, 256, 0, stream>>>(x, s);
    se_mlp_kernel<<<1, 32, 0, stream>>>(s, w1, w2, g1, b1, g2, b2, gate);
    se_scale_kernel<<<8192, 256, 0, stream>>>(x, gate, out);
}